// temporal_attention_32925219291318
// MI455X (gfx1250) — compile-verified
//
#include <hip/hip_runtime.h>

// Temporal attention for MI455X (gfx1250), fp32 path via V_WMMA_F32_16X16X4_F32.
// B=8 S=256 J=24 D=128 H=8 DEPTH=16

#define B_ 8
#define S_ 256
#define J_ 24
#define D_ 128
#define H_ 8
#define DEPTH_ 16
#define M_ (B_ * S_ * J_) /* 49152 rows in the flattened projection GEMMs */

#define QK_STRIDE 18   /* padded LDS row stride for 256x16 Q/K/V tiles  */
#define P_STRIDE 258   /* padded LDS row stride for 256x256 score matrix */

typedef __attribute__((ext_vector_type(2))) float v2f;
typedef __attribute__((ext_vector_type(8))) float v8f;

__device__ __forceinline__ v8f wmma_f32x4(v2f a, v2f b, v8f c) {
  // D = A(16x4 f32) * B(4x16 f32) + C(16x16 f32)
  return __builtin_amdgcn_wmma_f32_16x16x4_f32(
      /*neg_a=*/false, a, /*neg_b=*/false, b,
      /*c_mod=*/(short)0, c, /*reuse_a=*/false, /*reuse_b=*/false);
}

// ---------------------------------------------------------------------------
// Stage 0: xpe = x + positional_encoding(s, d)   (broadcast over b, j)
// ---------------------------------------------------------------------------
__global__ void __launch_bounds__(256) k_add_pe(const float* __restrict__ x,
                                                float* __restrict__ xpe) {
  int idx = blockIdx.x * 256 + threadIdx.x;   // grid sized exactly to M_*D_
  int d = idx & (D_ - 1);
  int s = (idx / (J_ * D_)) % S_;
  int i2 = d & ~1;
  // rate = 10000^(-i2/D) = exp(-ln(1e4) * i2 / D)
  float rate = __expf(-9.210340371976184f * (float)i2 * (1.0f / (float)D_));
  float ang = (float)s * rate;
  float pe = (d & 1) ? __cosf(ang) : __sinf(ang);
  xpe[idx] = x[idx] + pe;
}

// ---------------------------------------------------------------------------
// Stage 1/3: C[m,n] = sum_k A[m,k] * W[k,n] + bias[n]   (A: M_ x 128, W: 128x128)
// One block = 16-row stripe; 8 waves each own one 16-col tile; K=128 in 4-steps.
// ---------------------------------------------------------------------------
__global__ void __launch_bounds__(256) k_gemm128(const float* __restrict__ A,
                                                 const float* __restrict__ W,
                                                 const float* __restrict__ bias,
                                                 float* __restrict__ C) {
  __shared__ float as[16 * 130];  // padded stride 130 -> conflict-free frag loads
  const int m0 = blockIdx.x * 16;
  const int t = threadIdx.x;

  // Cooperative load of the 16x128 A stripe (coalesced).
#pragma unroll
  for (int blk = 0; blk < 8; ++blk) {
    int e = t + blk * 256;        // 0..2047
    int row = e >> 7, col = e & 127;
    as[row * 130 + col] = A[(m0 + row) * D_ + col];
  }
  __syncthreads();

  const int wave = t >> 5;        // 0..7 -> col tile
  const int lane = t & 31;
  const int lm = lane & 15;       // row-within-tile (A) / col-within-tile (B,D)
  const int lk = (lane >> 4) << 1;  // 0 or 2: which K-pair this half-wave holds
  const int n0 = wave * 16;

  v8f c = {};
  const float* wcol = W + n0 + lm;  // column n0+lm of W, row stride D_
#pragma unroll 4
  for (int kk = 0; kk < 128; kk += 4) {
    v2f a = *(const v2f*)&as[lm * 130 + kk + lk];       // A[m][kk+lk], A[m][kk+lk+1]
    v2f b;
    b.x = wcol[(kk + lk) * D_];                          // W[kk+lk][n]
    b.y = wcol[(kk + lk + 1) * D_];                      // W[kk+lk+1][n]
    c = wmma_f32x4(a, b, c);
  }

  const float bi = bias[n0 + lm];
#pragma unroll
  for (int r = 0; r < 8; ++r) {
    int row = m0 + r + ((lane >> 4) << 3);               // D layout: VGPR r -> M=r / r+8
    C[row * D_ + n0 + lm] = c[r] + bi;
  }
}

// ---------------------------------------------------------------------------
// Stage 2: per-head attention. One workgroup per (b, j, h).
// Whole head lives in LDS: Q/K/V (256x16, stride 18) + scores P (256x256, stride 258)
// = 319,488 B of the WGP's 320 KB.
// ---------------------------------------------------------------------------
__global__ void __launch_bounds__(256) k_attn(const float* __restrict__ q,
                                              const float* __restrict__ k,
                                              const float* __restrict__ v,
                                              const float* __restrict__ mask,
                                              float* __restrict__ out) {
  extern __shared__ float smem[];
  float* Qs = smem;                       // 256*18
  float* Ks = Qs + 256 * QK_STRIDE;       // 256*18
  float* Vs = Ks + 256 * QK_STRIDE;       // 256*18
  float* P  = Vs + 256 * QK_STRIDE;       // 256*258

  const int hid = blockIdx.x;             // 0..B*J*H-1
  const int h = hid & (H_ - 1);
  const int j = (hid / H_) % J_;
  const int b = hid / (H_ * J_);
  const int t = threadIdx.x;

  // Thread t loads row s=t of Q, K, V for this head (16 floats each, float4 x4).
  {
    long base = (((long)(b * S_ + t)) * J_ + j) * D_ + h * DEPTH_;
    const float4* q4 = (const float4*)(q + base);
    const float4* k4 = (const float4*)(k + base);
    const float4* v4 = (const float4*)(v + base);
#pragma unroll
    for (int c4 = 0; c4 < 4; ++c4) {
      float4 qv = q4[c4], kv = k4[c4], vv = v4[c4];
      int o = t * QK_STRIDE + c4 * 4;
      Qs[o + 0] = qv.x; Qs[o + 1] = qv.y; Qs[o + 2] = qv.z; Qs[o + 3] = qv.w;
      Ks[o + 0] = kv.x; Ks[o + 1] = kv.y; Ks[o + 2] = kv.z; Ks[o + 3] = kv.w;
      Vs[o + 0] = vv.x; Vs[o + 1] = vv.y; Vs[o + 2] = vv.z; Vs[o + 3] = vv.w;
    }
  }
  __syncthreads();

  const int wave = t >> 5;
  const int lane = t & 31;
  const int lm = lane & 15;
  const int lk = (lane >> 4) << 1;

  // ---- scores = Q * K^T / sqrt(DEPTH): 16x16 grid of 16x16 tiles, 8 waves ----
  for (int tile = wave; tile < 256; tile += 8) {
    const int q0 = (tile >> 4) * 16;
    const int n0 = (tile & 15) * 16;
    v8f c = {};
#pragma unroll
    for (int kk = 0; kk < DEPTH_; kk += 4) {
      v2f a = *(const v2f*)&Qs[(q0 + lm) * QK_STRIDE + kk + lk];
      v2f bf = *(const v2f*)&Ks[(n0 + lm) * QK_STRIDE + kk + lk]; // B[kd][n]=K[n][kd]
      c = wmma_f32x4(a, bf, c);
    }
#pragma unroll
    for (int r = 0; r < 8; ++r)
      P[(q0 + r + ((lane >> 4) << 3)) * P_STRIDE + n0 + lm] = c[r] * 0.25f;
  }
  __syncthreads();

  // ---- masked softmax, one thread per query row ----
  {
    const int qi = t;
    float* row = P + qi * P_STRIDE;
    float mx = -3.402823466e38f;
    if (qi >= 1) {
      const float* mrow = mask + (qi - 1) * (S_ - 1) - 1;  // mrow[ki]=mask[qi-1][ki-1]
      for (int ki = 0; ki < S_; ++ki) {
        float sv = row[ki];
        if (ki >= 1) sv += -1e9f * mrow[ki];   // only [1:,1:] block masked
        row[ki] = sv;
        mx = fmaxf(mx, sv);
      }
    } else {
      for (int ki = 0; ki < S_; ++ki) mx = fmaxf(mx, row[ki]);
    }
    float sum = 0.0f;
    for (int ki = 0; ki < S_; ++ki) {
      float e = __expf(row[ki] - mx);
      row[ki] = e;
      sum += e;
    }
    float inv = 1.0f / sum;
    for (int ki = 0; ki < S_; ++ki) row[ki] *= inv;
  }
  __syncthreads();

  // ---- out = P * V: 16 row-tiles, each wave does tiles {wave, wave+8} ----
#pragma unroll
  for (int tt = 0; tt < 2; ++tt) {
    const int q0 = (wave + tt * 8) * 16;
    v8f c = {};
    for (int kk = 0; kk < S_; kk += 4) {
      v2f a = *(const v2f*)&P[(q0 + lm) * P_STRIDE + kk + lk];
      v2f bf;
      bf.x = Vs[(kk + lk) * QK_STRIDE + lm];       // B[k][dd] = V[k][dd]
      bf.y = Vs[(kk + lk + 1) * QK_STRIDE + lm];
      c = wmma_f32x4(a, bf, c);
    }
#pragma unroll
    for (int r = 0; r < 8; ++r) {
      int srow = q0 + r + ((lane >> 4) << 3);
      long o = (((long)(b * S_ + srow)) * J_ + j) * D_ + h * DEPTH_ + lm;
      out[o] = c[r];   // scatter back to (B,S,J,D) layout for the Wo GEMM
    }
  }
}

// ---------------------------------------------------------------------------
extern "C" void kernel_launch(void* const* d_in, const int* in_sizes, int n_in,
                              void* d_out, int out_size, void* d_ws, size_t ws_size,
                              hipStream_t stream) {
  (void)in_sizes; (void)n_in; (void)out_size; (void)ws_size;
  const float* x    = (const float*)d_in[0];
  const float* mask = (const float*)d_in[1];
  const float* Wq   = (const float*)d_in[2];
  const float* bq   = (const float*)d_in[3];
  const float* Wk   = (const float*)d_in[4];
  const float* bk   = (const float*)d_in[5];
  const float* Wv   = (const float*)d_in[6];
  const float* bv   = (const float*)d_in[7];
  const float* Wo   = (const float*)d_in[8];
  const float* bo   = (const float*)d_in[9];
  float* out = (float*)d_out;

  float* ws = (float*)d_ws;
  const size_t nx = (size_t)M_ * D_;      // 6,291,456 floats per tensor
  float* xpe  = ws;                        // stage 0 result; reused as attn out
  float* qb   = ws + nx;
  float* kb   = ws + 2 * nx;
  float* vb   = ws + 3 * nx;
  float* attn = xpe;                       // xpe dead after V projection

  // Stage 0: positional encoding
  k_add_pe<<<(int)(nx / 256), 256, 0, stream>>>(x, xpe);

  // Stage 1: Q/K/V projections (fp32 WMMA GEMMs)
  k_gemm128<<<M_ / 16, 256, 0, stream>>>(xpe, Wq, bq, qb);
  k_gemm128<<<M_ / 16, 256, 0, stream>>>(xpe, Wk, bk, kb);
  k_gemm128<<<M_ / 16, 256, 0, stream>>>(xpe, Wv, bv, vb);

  // Stage 2: per-head attention, full head resident in 320KB WGP LDS
  const size_t smem = (size_t)(3 * 256 * QK_STRIDE + 256 * P_STRIDE) * sizeof(float);
  k_attn<<<B_ * J_ * H_, 256, smem, stream>>>(qb, kb, vb, mask, attn);

  // Stage 3: output projection straight into d_out
  k_gemm128<<<M_ / 16, 256, 0, stream>>>(attn, Wo, bo, out);

  // Tuple output: append the (unchanged) mask after the attention output.
  hipMemcpyAsync(out + nx, mask, (size_t)(S_ - 1) * (S_ - 1) * sizeof(float),
                 hipMemcpyDeviceToDevice, stream);
}